// GCN_53386443489915
// MI455X (gfx1250) — compile-verified
//
#include <hip/hip_runtime.h>
#include <hip/hip_bf16.h>

typedef float v2f __attribute__((ext_vector_type(2)));
typedef float v8f __attribute__((ext_vector_type(8)));

// -------------------- degree / normalization --------------------

__global__ void deg_init_k(float* __restrict__ deg, int n) {
    int i = blockIdx.x * blockDim.x + threadIdx.x;
    if (i < n) deg[i] = 2.0f;   // improved=True self-loop weight
}

__global__ void deg_acc_k(const int* __restrict__ col, float* __restrict__ deg, int e) {
    int i = blockIdx.x * blockDim.x + threadIdx.x;
    if (i < e) atomicAdd(&deg[col[i]], 1.0f);
}

__global__ void dinv_k(float* __restrict__ deg, int n) {
    int i = blockIdx.x * blockDim.x + threadIdx.x;
    if (i < n) {
        float d = deg[i];
        deg[i] = (d > 0.0f) ? rsqrtf(d) : 0.0f;
    }
}

__global__ void selfnorm_k(const float* __restrict__ dinv, float* __restrict__ sn, int n) {
    int i = blockIdx.x * blockDim.x + threadIdx.x;
    if (i < n) {
        float v = dinv[i];
        sn[i] = 2.0f * v * v;   // weight 2 self-loop
    }
}

__global__ void edgenorm_k(const int* __restrict__ row, const int* __restrict__ col,
                           const float* __restrict__ dinv, float* __restrict__ nrm, int e) {
    int i = blockIdx.x * blockDim.x + threadIdx.x;
    if (i < e) nrm[i] = dinv[row[i]] * dinv[col[i]];
}

__global__ void zero4_k(float4* __restrict__ p, long long n4) {
    long long i = (long long)blockIdx.x * blockDim.x + threadIdx.x;
    if (i < n4) p[i] = make_float4(0.f, 0.f, 0.f, 0.f);
}

// -------------------- weight pre-pack --------------------
// Wp[k2][c][j] = W[2*k2+j][c] for c < M else 0;  shape (K/2, Mp, 2), Mp % 64 == 0.
// Gives each WMMA lane a contiguous v2f B-fragment load with no bounds check.
__global__ void packW_k(const float* __restrict__ W, float* __restrict__ Wp,
                        int K, int M, int Mp) {
    int idx = blockIdx.x * blockDim.x + threadIdx.x;
    int total = (K >> 1) * Mp;
    if (idx >= total) return;
    int k2 = idx / Mp;
    int c  = idx % Mp;
    float v0 = 0.f, v1 = 0.f;
    if (c < M) {
        v0 = W[(size_t)(2 * k2)     * M + c];
        v1 = W[(size_t)(2 * k2 + 1) * M + c];
    }
    float* dst = Wp + ((size_t)k2 * Mp + c) * 2;
    dst[0] = v0;
    dst[1] = v1;
}

// -------------------- edge scatter-add --------------------
// ACC[col[e], :] += nrm[e] * H[row[e], :]
__global__ void scatter_k(const float* __restrict__ H, const int* __restrict__ row,
                          const int* __restrict__ col, const float* __restrict__ nrm,
                          float* __restrict__ ACC, int E, int M) {
    int e = blockIdx.x * blockDim.x + threadIdx.x;
    if (e >= E) return;
    float w = nrm[e];
    const float4* src = (const float4*)(H + (size_t)row[e] * M);
    float* dst = ACC + (size_t)col[e] * M;
    int m4 = M >> 2;
    for (int f = 0; f < m4; ++f) {
        float4 h = src[f];
        float* d = dst + (f << 2);
        atomicAdd(d + 0, w * h.x);
        atomicAdd(d + 1, w * h.y);
        atomicAdd(d + 2, w * h.z);
        atomicAdd(d + 3, w * h.w);
    }
}

// -------------------- finalize: self-loop + bias + relu --------------------
__global__ void finalize_k(const float* __restrict__ ACC, const float* __restrict__ SRC,
                           const float* __restrict__ sn, const float* __restrict__ bias,
                           float* __restrict__ OUT, int M, int hasBias, int relu) {
    int i = blockIdx.x;
    int j = threadIdx.x;
    if (j >= M) return;
    size_t idx = (size_t)i * M + j;
    float v = ACC[idx] + sn[i] * SRC[idx];
    if (hasBias) v += bias[j];
    if (relu) v = fmaxf(v, 0.0f);
    OUT[idx] = v;
}

// -------------------- WMMA fp32 GEMM: H = X @ W (+bias)(+relu) --------------------
// One wave computes a 16x64 output strip with 4x V_WMMA_F32_16X16X4_F32.
// Wp is pair-packed (K/2, Mp, 2) with Mp % 64 == 0 -> unguarded b64 B-loads.
// epilogue: 0 = none, 2 = +bias then relu
__global__ void __launch_bounds__(128)
gemm_wmma_k(const float* __restrict__ X, const float* __restrict__ Wp,
            const float* __restrict__ bias, float* __restrict__ H,
            int N, int K, int M, int Mp, int epilogue) {
    const int lane     = threadIdx.x & 31;
    const int wave     = blockIdx.x * 4 + (threadIdx.x >> 5);
    const int colQuads = Mp >> 6;
    const int rowTiles = (N + 15) >> 4;
    const int rowTile  = wave / colQuads;
    const int cq       = wave % colQuads;
    if (rowTile >= rowTiles) return;      // wave-uniform exit; EXEC all-ones below

    const int cl    = lane & 15;
    const int hi    = lane >> 4;          // 0 or 1
    const int cbase = cq << 6;
    const size_t stride = (size_t)Mp * 2; // floats per packed k-pair row

    v8f acc[4] = {};
    const float* __restrict__ aptr = X + (size_t)(rowTile * 16 + cl) * K + (hi << 1);
    const float* __restrict__ wptr = Wp + (size_t)hi * stride + (size_t)(cbase + cl) * 2;

    for (int k = 0; k < K; k += 4) {
        v2f a = *(const v2f*)(aptr);      // A: rows striped over lanes, K pair by lane[4]
        v2f b0 = *(const v2f*)(wptr);     // B: col = cbase + t*16 + cl, packed k-pair
        v2f b1 = *(const v2f*)(wptr + 32);
        v2f b2 = *(const v2f*)(wptr + 64);
        v2f b3 = *(const v2f*)(wptr + 96);
        acc[0] = __builtin_amdgcn_wmma_f32_16x16x4_f32(false, a, false, b0, (short)0, acc[0], false, false);
        acc[1] = __builtin_amdgcn_wmma_f32_16x16x4_f32(false, a, false, b1, (short)0, acc[1], false, false);
        acc[2] = __builtin_amdgcn_wmma_f32_16x16x4_f32(false, a, false, b2, (short)0, acc[2], false, false);
        acc[3] = __builtin_amdgcn_wmma_f32_16x16x4_f32(false, a, false, b3, (short)0, acc[3], false, false);
        aptr += 4;
        wptr += 2 * stride;
    }

    // D layout: VGPR v -> row rbase+v; col = cbase + t*16 + cl
    const int rbase = rowTile * 16 + (hi << 3);
#pragma unroll
    for (int t = 0; t < 4; ++t) {
        int c = cbase + t * 16 + cl;
        if (c >= M) continue;             // only live when M==40 (padded to 64)
        float bv = (epilogue > 0) ? bias[c] : 0.0f;
#pragma unroll
        for (int v = 0; v < 8; ++v) {
            int rr = rbase + v;
            if (rr >= N) continue;
            float val = acc[t][v] + bv;
            if (epilogue == 2) val = fmaxf(val, 0.0f);
            H[(size_t)rr * M + c] = val;
        }
    }
}

// -------------------- host orchestration --------------------

static inline int cdiv(long long a, long long b) { return (int)((a + b - 1) / b); }

extern "C" void kernel_launch(void* const* d_in, const int* in_sizes, int n_in,
                              void* d_out, int out_size, void* d_ws, size_t ws_size,
                              hipStream_t stream) {
    const float* x  = (const float*)d_in[0];
    const int*   ei = (const int*)  d_in[1];
    const float* W1 = (const float*)d_in[2];
    const float* b1 = (const float*)d_in[3];
    const float* W2 = (const float*)d_in[4];
    const float* b2 = (const float*)d_in[5];
    const float* W3 = (const float*)d_in[6];
    const float* b3 = (const float*)d_in[7];
    const float* W4 = (const float*)d_in[8];
    const float* b4 = (const float*)d_in[9];

    const int N = in_sizes[0] / 128;
    const int E = in_sizes[1] / 2;
    const int* row = ei;
    const int* col = ei + E;

    float* ws   = (float*)d_ws;
    float* dinv = ws;  ws += N;                 // deg, then rsqrt in-place
    float* sn   = ws;  ws += N;                 // self-loop norm
    float* nrm  = ws;  ws += E;                 // per-edge norm
    float* Wp1  = ws;  ws += 128 * 256;         // packed weights
    float* Wp2  = ws;  ws += 256 * 128;
    float* Wp3  = ws;  ws += 128 * 64;
    float* Wp4  = ws;  ws += 64 * 64;           // padded 40 -> 64 cols
    float* B0   = ws;  ws += (size_t)N * 128;   // GEMM scratch (layers 2-4)
    float* B1   = ws;  ws += (size_t)N * 128;   // ping
    float* B2   = ws;  /* N*256 */              // pong (256-wide act1)

    auto pack = [&](const float* Wsrc, float* Wdst, int K, int M, int Mp) {
        int total = (K / 2) * Mp;
        packW_k<<<cdiv(total, 256), 256, 0, stream>>>(Wsrc, Wdst, K, M, Mp);
    };
    auto gemm = [&](const float* Xp, const float* Wpk, const float* bp, float* Hp,
                    int K, int M, int Mp, int epi) {
        long long waves = (long long)((N + 15) / 16) * (Mp / 64);
        gemm_wmma_k<<<cdiv(waves, 4), 128, 0, stream>>>(Xp, Wpk, bp, Hp, N, K, M, Mp, epi);
    };
    auto zero = [&](float* p, long long nfloats) {
        long long n4 = nfloats >> 2;
        zero4_k<<<cdiv(n4, 256), 256, 0, stream>>>((float4*)p, n4);
    };
    auto scatter = [&](const float* Hp, float* ACC, int M) {
        scatter_k<<<cdiv(E, 256), 256, 0, stream>>>(Hp, row, col, nrm, ACC, E, M);
    };
    auto finalize = [&](const float* ACC, const float* SRC, const float* bp,
                        float* OUT, int M, int relu) {
        int bdim = ((M + 31) / 32) * 32;
        finalize_k<<<N, bdim, 0, stream>>>(ACC, SRC, sn, bp, OUT, M, bp != nullptr, relu);
    };

    // ---- gcn_norm (improved=True) + weight packing ----
    deg_init_k<<<cdiv(N, 256), 256, 0, stream>>>(dinv, N);
    deg_acc_k <<<cdiv(E, 256), 256, 0, stream>>>(col, dinv, E);
    dinv_k    <<<cdiv(N, 256), 256, 0, stream>>>(dinv, N);
    selfnorm_k<<<cdiv(N, 256), 256, 0, stream>>>(dinv, sn, N);
    edgenorm_k<<<cdiv(E, 256), 256, 0, stream>>>(row, col, dinv, nrm, E);
    pack(W1, Wp1, 128, 256, 256);
    pack(W2, Wp2, 256, 128, 128);
    pack(W3, Wp3, 128,  64,  64);
    pack(W4, Wp4,  64,  40,  64);

    // ---- Layer 1 (aggregate-first: scatter on 128-wide input, not 256-wide out) ----
    zero(B1, (long long)N * 128);
    scatter(x, B1, 128);
    finalize(B1, x, nullptr, B1, 128, 0);               // B1 = Agg(X)
    gemm(B1, Wp1, b1, B2, 128, 256, 256, /*bias+relu*/2); // B2 = relu(Agg(X)@W1+b1)

    // ---- Layer 2 (transform-first) ----
    gemm(B2, Wp2, nullptr, B0, 256, 128, 128, 0);       // B0 = act1 @ W2
    zero(B1, (long long)N * 128);
    scatter(B0, B1, 128);
    finalize(B1, B0, b2, B1, 128, /*relu*/1);           // B1 = act2

    // ---- Layer 3 ----
    gemm(B1, Wp3, nullptr, B0, 128, 64, 64, 0);         // B0 = act2 @ W3
    zero(B2, (long long)N * 64);
    scatter(B0, B2, 64);
    finalize(B2, B0, b3, B2, 64, 0);                    // B2 = act3

    // ---- Layer 4 (M=40, W padded to 64 cols; stores guarded) ----
    gemm(B2, Wp4, nullptr, B0, 64, 40, 64, 0);          // B0 = act3 @ W4
    zero(B1, (long long)N * 40);
    scatter(B0, B1, 40);
    finalize(B1, B0, b4, (float*)d_out, 40, 0);         // d_out = final
}